// ModalVerlet_68118181314859
// MI455X (gfx1250) — compile-verified
//
#include <hip/hip_runtime.h>

// Problem constants (from reference): B=16, M=64, T=48000, fs=48000
#define NB 16
#define NM 64
#define NT 48000
#define KDT  (1.0f / 48000.0f)
#define HKDT (0.5f / 48000.0f)
#define TILE 16
#define NTILES (NT / TILE)   // 3000 tiles cover t = 0..47999 (tile n: steps s = 16n-1+i, i where valid)
#define PITCH 20             // floats per staged channel row: 80B (16B aligned), <=2-way LDS bank conflicts

typedef __attribute__((ext_vector_type(2))) float v2f;
typedef __attribute__((ext_vector_type(8))) float v8f;
typedef int v4i __attribute__((vector_size(4 * sizeof(int))));  // matches builtin param type

#if __has_builtin(__builtin_amdgcn_wmma_f32_16x16x4_f32)
#define HAVE_WMMA_F32X4 1
#else
#define HAVE_WMMA_F32X4 0
#warning "gfx1250: __builtin_amdgcn_wmma_f32_16x16x4_f32 not available; using scalar w-reduction fallback"
#endif

#if __has_builtin(__builtin_amdgcn_global_store_async_from_lds_b128)
#define HAVE_ASYNC_ST 1
#else
#define HAVE_ASYNC_ST 0
#warning "gfx1250: __builtin_amdgcn_global_store_async_from_lds_b128 not available; using sync b128 writeback"
#endif

__device__ __forceinline__ float ftanh(float x) {
#if __has_builtin(__builtin_amdgcn_tanhf)
  return __builtin_amdgcn_tanhf(x);
#else
  float r;
  asm("v_tanh_f32 %0, %1" : "=v"(r) : "v"(x));
  return r;
#endif
}

__device__ __forceinline__ float bcast(float v, int lane) {
  return __builtin_bit_cast(float, __builtin_amdgcn_readlane(__builtin_bit_cast(int, v), lane));
}

__device__ __forceinline__ void wait_dscnt0() {
#if __has_builtin(__builtin_amdgcn_s_wait_dscnt)
  __builtin_amdgcn_s_wait_dscnt(0);
#else
  asm volatile("s_wait_dscnt 0" ::: "memory");
#endif
}

template <int N>
__device__ __forceinline__ void wait_asynccnt() {
#if __has_builtin(__builtin_amdgcn_s_wait_asynccnt)
  __builtin_amdgcn_s_wait_asynccnt(N);
#else
  asm volatile("s_wait_asynccnt %0" ::"i"(N) : "memory");
#endif
}

__device__ __forceinline__ void store_async_b128(float* g, const float* l) {
#if HAVE_ASYNC_ST
  __builtin_amdgcn_global_store_async_from_lds_b128(
      (__attribute__((address_space(1))) v4i*)g,
      (__attribute__((address_space(3))) v4i*)l, 0, 0);
#else
  *(float4*)g = *(const float4*)l;  // both 16B aligned by construction
#endif
}

__global__ __launch_bounds__(32, 1) void ModalVerlet_kernel(
    const float* __restrict__ y0, const float* __restrict__ om,
    const float* __restrict__ sg, const float* __restrict__ gm,
    const float* __restrict__ phie, const float* __restrict__ phio,
    const float* __restrict__ fe, float* __restrict__ y, float* __restrict__ w) {
  const int b = blockIdx.x;   // one wave per batch
  const int l = threadIdx.x;  // lane 0..31; owns modes l and l+32

  __shared__ __align__(16) float stage[2][128 * PITCH];  // [chan 0..127][t-in-tile], double buffered
  __shared__ float red[16];                              // w repack

  const int ma = l, mb = l + 32;
  float om2a = om[b * NM + ma]; om2a *= om2a;
  float om2b = om[b * NM + mb]; om2b *= om2b;
  const float sga = sg[b * NM + ma], sgb = sg[b * NM + mb];
  const float s2a = 2.0f * sga, s2b = 2.0f * sgb;
  const float invda = 1.0f / (1.0f + KDT * sga);
  const float invdb = 1.0f / (1.0f + KDT * sgb);
  const float pea = phie[b * NM + ma], peb = phie[b * NM + mb];
  const float g = gm[b];
  const float g2 = g * g;

  float qa = y0[b * 2 * NM + ma], qb = y0[b * 2 * NM + mb];
  float pa = y0[b * 2 * NM + NM + ma], pb = y0[b * 2 * NM + NM + mb];
  float ta = ftanh(qa), tb = ftanh(qb);

  // ---- Constant WMMA B operand: B(4x16) f32, column N=0 holds Phi_o chunk [4j..4j+3].
  // 32-bit B layout (mirrors A/C conventions): VGPR0: K=0 (lanes0-15) / K=2 (lanes16-31); VGPR1: K=1 / K=3.
  const int kh = (l >> 4) * 2;         // K sub-pair this lane-half supplies
  const bool bact = (l & 15) == 0;     // only N==0 column is nonzero
  v2f Bj[16];
#pragma unroll
  for (int j = 0; j < 16; ++j) {
    Bj[j].x = bact ? phio[b * NM + 4 * j + kh] : 0.0f;
    Bj[j].y = bact ? phio[b * NM + 4 * j + kh + 1] : 0.0f;
  }

  float* __restrict__ yb = y + (size_t)b * (2 * NM) * NT;
  float* __restrict__ wb = w + (size_t)b * NT;
  const float* __restrict__ feb = fe + (size_t)b * NT;
  const int row = l & 15;
  const int csub = l >> 2, grp = l & 3;

  for (int n = 0; n < NTILES; ++n) {
    wait_asynccnt<16>();  // tile n-2's 16 async stores done -> buffer (n&1) reusable
    float* st = stage[n & 1];
    const int t0 = n * TILE;

    // 17 fe values needed: fe[t0-1 .. t0+15]; clamped coalesced load, v_readlane broadcast.
    int fidx = t0 - 1 + l;
    fidx = fidx < 0 ? 0 : (fidx > NT - 1 ? NT - 1 : fidx);
    const float fev = feb[fidx];

#pragma unroll
    for (int i = 0; i < TILE; ++i) {
      if (i != 0 || n != 0) {  // tile 0 / col 0 is the initial condition y0
        const float fe0 = bcast(fev, i);
        const float fe1 = bcast(fev, i + 1);
        // mode a
        float pd = fmaf(fe0, pea, fmaf(g2, ta, fmaf(-om2a, qa, -s2a * pa)));
        float ph = fmaf(HKDT, pd, pa);
        qa = fmaf(KDT, ph, qa);
        ta = ftanh(qa);
        float vv = fmaf(-om2a, qa, fmaf(g2, ta, fe1 * pea));
        pa = fmaf(HKDT, vv, ph) * invda;
        // mode b (independent -> pipelines with mode a)
        pd = fmaf(fe0, peb, fmaf(g2, tb, fmaf(-om2b, qb, -s2b * pb)));
        ph = fmaf(HKDT, pd, pb);
        qb = fmaf(KDT, ph, qb);
        tb = ftanh(qb);
        vv = fmaf(-om2b, qb, fmaf(g2, tb, fe1 * peb));
        pb = fmaf(HKDT, vv, ph) * invdb;
      }
      // stage state for t = t0 + i  (channels: q -> c=m, p -> c=64+m)
      st[ma * PITCH + i] = qa;
      st[mb * PITCH + i] = qb;
      st[(NM + ma) * PITCH + i] = pa;
      st[(NM + mb) * PITCH + i] = pb;
    }
    wait_dscnt0();  // staged tile visible cross-lane (WMMA A loads) and to async DMA engine

    // ---- w[t0 .. t0+15] = Q(16x64) * Phi_o via 16 chained V_WMMA_F32_16X16X4_F32
#if HAVE_WMMA_F32X4
    {
      v8f acc = {};
#pragma unroll
      for (int j = 0; j < 16; ++j) {
        const int c0 = 4 * j + kh;
        v2f a;  // A 16x4 f32: lanes0-15 rows (=t), VGPR0=K{0|2}, VGPR1=K{1|3}
        a.x = st[c0 * PITCH + row];
        a.y = st[(c0 + 1) * PITCH + row];
        acc = __builtin_amdgcn_wmma_f32_16x16x4_f32(false, a, false, Bj[j],
                                                    (short)0, acc, false, false);
      }
      // D column N=0 lives in lanes 0 (M=r) and 16 (M=r+8)
      if (bact) {
        const int h8 = (l >> 4) * 8;
#pragma unroll
        for (int r = 0; r < 8; ++r) red[h8 + r] = acc[r];
      }
    }
#else
    if (l < 16) {
      float s = 0.0f;
      for (int m = 0; m < NM; ++m) s = fmaf(phio[b * NM + m], st[m * PITCH + l], s);
      red[l] = s;
    }
#endif
    wait_dscnt0();
    if (l < 16) wb[t0 + l] = red[l];  // coalesced 16-float w store

    // ---- y writeback: 128 channels x 16 timesteps, 16 x async b128 (DMA, off critical path)
#pragma unroll
    for (int it = 0; it < 16; ++it) {
      const int cch = it * 8 + csub;  // 8 channels x 4 groups-of-4-t per instruction
      store_async_b128(yb + (size_t)cch * NT + t0 + grp * 4,
                       st + cch * PITCH + grp * 4);
    }
  }
  wait_asynccnt<0>();
}

extern "C" void kernel_launch(void* const* d_in, const int* in_sizes, int n_in,
                              void* d_out, int out_size, void* d_ws, size_t ws_size,
                              hipStream_t stream) {
  // setup_inputs order: fs, num_samples, y0, omega, sigma, gamma, Phi_e, Phi_o, fe_points
  const float* y0   = (const float*)d_in[2];
  const float* om   = (const float*)d_in[3];
  const float* sg   = (const float*)d_in[4];
  const float* gm   = (const float*)d_in[5];
  const float* phie = (const float*)d_in[6];
  const float* phio = (const float*)d_in[7];
  const float* fe   = (const float*)d_in[8];
  float* y = (float*)d_out;
  float* w = y + (size_t)NB * 2 * NM * NT;  // tuple (y, w) concatenated flat
  ModalVerlet_kernel<<<dim3(NB), dim3(32), 0, stream>>>(y0, om, sg, gm, phie, phio, fe, y, w);
}